// DenseRoutingMaskLayer_56624848830879
// MI455X (gfx1250) — compile-verified
//
#include <hip/hip_runtime.h>

#define ROUTES        8
#define B_ROWS        16384
#define D_COLS        4096
#define CHUNK         (D_COLS / ROUTES)   // 512 floats per route chunk
#define CHUNK4        (CHUNK / 4)         // 128 float4 per chunk
#define ROWS_PER_BLK  8
#define THREADS       256                 // 8 wave32 -> 8 rows per block

#if defined(__AMDGCN__) && \
    __has_builtin(__builtin_amdgcn_global_load_async_to_lds_b128) && \
    __has_builtin(__builtin_amdgcn_global_store_async_from_lds_b128)
#define USE_ASYNC_LDS 1
#else
#define USE_ASYNC_LDS 0
#endif

// Builtin-matching pointer types (from hipcc diagnostic: param is
// 'int __vector_size__(16) *' — global side AS(1), LDS side AS(3)).
typedef int v4i __attribute__((__vector_size__(16)));
typedef __attribute__((address_space(1))) v4i g_v4i;
typedef __attribute__((address_space(3))) v4i l_v4i;

__global__ __launch_bounds__(THREADS)
void route_gather_kernel(const float* __restrict__ inputs,
                         const float* __restrict__ routing,
                         float* __restrict__ out)
{
    const int wave = threadIdx.x >> 5;          // 0..7, one row per wave32
    const int lane = threadIdx.x & 31;
    const int row  = blockIdx.x * ROWS_PER_BLK + wave;

    // ---- argmax over 8 routing logits (first-max wins, like jnp.argmax) ----
    const float4* r4 = reinterpret_cast<const float4*>(routing + (size_t)row * ROUTES);
    float4 ra = r4[0];
    float4 rb = r4[1];
    float rv[ROUTES] = {ra.x, ra.y, ra.z, ra.w, rb.x, rb.y, rb.z, rb.w};
    int   best = 0;
    float bv   = rv[0];
#pragma unroll
    for (int i = 1; i < ROUTES; ++i) {
        if (rv[i] > bv) { bv = rv[i]; best = i; }
    }
    // Wave-uniform by construction; force into an SGPR so the chunk base is scalar.
    best = __builtin_amdgcn_readfirstlane(best);

    const float4* src4 = reinterpret_cast<const float4*>(inputs)
                         + (size_t)row * (D_COLS / 4) + (size_t)best * CHUNK4;
    float4* dst4 = reinterpret_cast<float4*>(out) + (size_t)row * CHUNK4;

#if USE_ASYNC_LDS
    // CDNA5 async data-mover path: global -> LDS -> global, b128 packets,
    // zero VGPRs spent on payload. Each wave owns a private 2 KB LDS slice.
    __shared__ float4 stage[ROWS_PER_BLK * CHUNK4];   // 16 KB per block
    l_v4i* l4 = (l_v4i*)stage + (size_t)wave * CHUNK4;

#pragma unroll
    for (int j = 0; j < 4; ++j) {
        __builtin_amdgcn_global_load_async_to_lds_b128(
            (g_v4i*)(src4 + j * 32 + lane),
            (l_v4i*)(l4 + j * 32 + lane),
            /*offset=*/0, /*cpol=*/0);
    }

#if __has_builtin(__builtin_amdgcn_s_wait_asynccnt)
    __builtin_amdgcn_s_wait_asynccnt(0);
#else
    asm volatile("s_wait_asynccnt 0" ::: "memory");
#endif

#pragma unroll
    for (int j = 0; j < 4; ++j) {
        __builtin_amdgcn_global_store_async_from_lds_b128(
            (g_v4i*)(dst4 + j * 32 + lane),
            (l_v4i*)(l4 + j * 32 + lane),
            /*offset=*/0, /*cpol=*/0);
    }
    // S_ENDPGM performs an implicit wait-idle; stores drain before wave retire.
#else
    // Fallback: direct VGPR b128 copy (4 independent loads in flight per lane).
    float4 v0 = src4[lane +  0];
    float4 v1 = src4[lane + 32];
    float4 v2 = src4[lane + 64];
    float4 v3 = src4[lane + 96];
    dst4[lane +  0] = v0;
    dst4[lane + 32] = v1;
    dst4[lane + 64] = v2;
    dst4[lane + 96] = v3;
#endif
}

extern "C" void kernel_launch(void* const* d_in, const int* in_sizes, int n_in,
                              void* d_out, int out_size, void* d_ws, size_t ws_size,
                              hipStream_t stream) {
    const float* inputs  = (const float*)d_in[0];   // [16384, 4096] f32
    const float* routing = (const float*)d_in[1];   // [16384, 8]    f32
    float*       out     = (float*)d_out;           // [16384, 512]  f32

    (void)in_sizes; (void)n_in; (void)out_size; (void)d_ws; (void)ws_size;

    const int grid = B_ROWS / ROWS_PER_BLK;         // 2048 blocks
    route_gather_kernel<<<grid, THREADS, 0, stream>>>(inputs, routing, out);
}